// MovingAverage_28905129902258
// MI455X (gfx1250) — compile-verified
//
#include <hip/hip_runtime.h>
#include <stdint.h>

// ---- problem constants (match reference) ----
#define GAP     64
#define NROWS   4096
#define LCOLS   8192
#define OUTC    (LCOLS - GAP + 1)   // 8129

// ---- tiling ----
#define MT        16                 // rows per block (one WMMA M)
#define COLS_PB   512                // output columns per block
#define NTILES    (COLS_PB / 16)     // 32 16-wide output tiles per block
#define KWIN      (16 + GAP)         // 80 input cols feeding one 16-wide tile
#define KCHUNKS   (KWIN / 4)         // 20 WMMA f32 16x16x4 steps per tile
#define LDS_STRIDE 580               // floats; 2320B rows: 16B aligned, conflict-free
#define STAGE_F    (COLS_PB + GAP)   // 576 floats staged per row
#define THREADS    256               // 8 wave32s

typedef __attribute__((ext_vector_type(2))) float v2f;
typedef __attribute__((ext_vector_type(4))) float v4f;
typedef __attribute__((ext_vector_type(8))) float v8f;

// exact pointee type the async builtin expects: int __vector_size__(16)
typedef int v4i __attribute__((__vector_size__(16)));
typedef v4i __attribute__((address_space(1))) gas_v4i;   // global
typedef v4i __attribute__((address_space(3))) las_v4i;   // LDS

#if defined(__has_builtin)
#if __has_builtin(__builtin_amdgcn_global_load_async_to_lds_b128)
#define HAVE_ASYNC_LDS 1
#endif
#if __has_builtin(__builtin_amdgcn_s_wait_asynccnt)
#define HAVE_WAIT_ASYNC 1
#endif
#endif

__global__ __launch_bounds__(THREADS)
void MovingAverage_wmma_kernel(const float* __restrict__ in,
                               float* __restrict__ out) {
    __shared__ float lds[MT * LDS_STRIDE];

    const int cb  = blockIdx.x * COLS_PB;   // output-column base of this block
    const int rb  = blockIdx.y * MT;        // row base of this block
    const int tid = threadIdx.x;

    // ---------------- Stage 16 rows x 576 cols into LDS ----------------
    // 144 b128 chunks per row * 16 rows = 2304 chunks; 256 threads -> 9 each.
    // Zero-fill chunks whose global column >= LCOLS (tail block halo).
    #pragma unroll
    for (int it = 0; it < (MT * (STAGE_F / 4)) / THREADS; ++it) {
        const int idx  = tid + it * THREADS;
        const int r    = idx / (STAGE_F / 4);
        const int c4   = (idx % (STAGE_F / 4)) * 4;
        const int gcol = cb + c4;
        float* lp = &lds[r * LDS_STRIDE + c4];
        if (gcol < LCOLS) {
            const float* gp = in + (size_t)(rb + r) * LCOLS + gcol;
#if defined(HAVE_ASYNC_LDS)
            // async DMA: global -> LDS, 16B per lane, tracked by ASYNCcnt
            __builtin_amdgcn_global_load_async_to_lds_b128(
                (gas_v4i*)(uintptr_t)gp,
                (las_v4i*)(uint32_t)(uintptr_t)lp,
                0, 0);
#else
            *(v4f*)lp = *(const v4f*)gp;
#endif
        } else {
            v4f z = {0.f, 0.f, 0.f, 0.f};
            *(v4f*)lp = z;
        }
    }
#if defined(HAVE_ASYNC_LDS)
#if defined(HAVE_WAIT_ASYNC)
    __builtin_amdgcn_s_wait_asynccnt(0);
#else
    asm volatile("s_wait_asynccnt 0" ::: "memory");
#endif
#endif
    __syncthreads();

    // ---------------- Weight fragments (banded box matrix / 64) ----------------
    // B is 4x16 (KxN) per chunk: VGPR0 = rows K0 (lanes 0-15) / K2 (lanes 16-31),
    //                            VGPR1 = rows K1 / K3.  W[u,t] = (t<=u<t+64)/64.
    const int lane  = tid & 31;
    const int lhalf = lane >> 4;   // 0: lanes 0-15, 1: lanes 16-31
    const int ln    = lane & 15;   // N (output column within tile) / M for A

    v2f Bfrag[KCHUNKS];
    #pragma unroll
    for (int p = 0; p < KCHUNKS; ++p) {
        const int k0 = 4 * p + 2 * lhalf;
        Bfrag[p].x = (k0     >= ln && k0     < ln + GAP) ? (1.0f / GAP) : 0.0f;
        Bfrag[p].y = (k0 + 1 >= ln && k0 + 1 < ln + GAP) ? (1.0f / GAP) : 0.0f;
    }

    // ---------------- WMMA tiles: one 16x16 output tile per wave per step ----------------
    const int wave = tid >> 5;     // 8 waves, each does NTILES/8 = 4 tiles
    for (int t = wave; t < NTILES; t += THREADS / 32) {
        const int tilebase = t * 16;            // block-local output column base
        v8f c = {};                             // f32 accumulator, starts at 0
        #pragma unroll
        for (int p = 0; p < KCHUNKS; ++p) {
            // A 16x4 f32 fragment: lane holds row ln, K = 4p + 2*lhalf + {0,1}
            const int acol = tilebase + 4 * p + 2 * lhalf;
            v2f a = *(const v2f*)&lds[ln * LDS_STRIDE + acol];
            c = __builtin_amdgcn_wmma_f32_16x16x4_f32(
                    /*neg_a=*/false, a, /*neg_b=*/false, Bfrag[p],
                    /*c_mod=*/(short)0, c, /*reuse_a=*/false, /*reuse_b=*/false);
        }
        // D layout: VGPR i -> row (i + 8*lhalf), column = ln (per lane half)
        const int ocol = cb + tilebase + ln;
        if (ocol < OUTC) {
            float* op = out + (size_t)(rb + 8 * lhalf) * OUTC + ocol;
            #pragma unroll
            for (int i = 0; i < 8; ++i)
                op[(size_t)i * OUTC] = c[i];
        }
    }
}

extern "C" void kernel_launch(void* const* d_in, const int* in_sizes, int n_in,
                              void* d_out, int out_size, void* d_ws, size_t ws_size,
                              hipStream_t stream) {
    (void)in_sizes; (void)n_in; (void)out_size; (void)d_ws; (void)ws_size;
    const float* in = (const float*)d_in[0];
    float* out      = (float*)d_out;

    dim3 grid((OUTC + COLS_PB - 1) / COLS_PB,   // 16 column blocks
              NROWS / MT);                      // 256 row blocks
    MovingAverage_wmma_kernel<<<grid, THREADS, 0, stream>>>(in, out);
}